// LongTermMemory_47906065220009
// MI455X (gfx1250) — compile-verified
//
#include <hip/hip_runtime.h>
#include <math.h>

#define BB 32
#define NN 8192
#define DD 64
#define KSPLIT 4
#define KCH (NN / KSPLIT)   // 2048 K-iterations per wave section

typedef float v2f __attribute__((ext_vector_type(2)));
typedef float v8f __attribute__((ext_vector_type(8)));

// ---------------------------------------------------------------------------
// K1: sequential erase/add scan over batch for each row of M, plus row 1/norm.
// One block per row n (64 threads = D).
// ---------------------------------------------------------------------------
__global__ __launch_bounds__(64) void k_write_scan(
    const float* __restrict__ wt, const float* __restrict__ e,
    const float* __restrict__ a, const float* __restrict__ M,
    float* __restrict__ Mf, float* __restrict__ invnorm)
{
  const int n = blockIdx.x;
  const int d = threadIdx.x;
  float m = M[n * DD + d];
  for (int b = 0; b < BB; ++b) {
    const float wb = wt[b * NN + n];
    m = m * (1.0f - wb * e[b * DD + d]) + wb * a[b * DD + d];
  }
  Mf[n * DD + d] = m;
  float ss = m * m;
  #pragma unroll
  for (int off = 16; off > 0; off >>= 1) ss += __shfl_down(ss, off);
  __shared__ float sh[2];
  if ((threadIdx.x & 31) == 0) sh[threadIdx.x >> 5] = ss;
  __syncthreads();
  if (threadIdx.x == 0) {
    const float nrm = sqrtf(sh[0] + sh[1]);
    invnorm[n] = 1.0f / fmaxf(nrm, 1e-8f);
  }
}

// ---------------------------------------------------------------------------
// K2: cosine-similarity addressing + stable softmax -> w[b][n].
// One block per batch row b (256 threads).
// ---------------------------------------------------------------------------
__global__ __launch_bounds__(256) void k_address(
    const float* __restrict__ key, const float* __restrict__ Mf,
    const float* __restrict__ invnorm, float* __restrict__ sim,
    float* __restrict__ w)
{
  const int b = blockIdx.x;
  const int t = threadIdx.x;
  __shared__ float kn[DD];
  __shared__ float red[8];
  __shared__ float bc;

  if (t == 0) {
    float s = 0.f;
    for (int dd = 0; dd < DD; ++dd) { const float kv = key[b*DD+dd]; s += kv*kv; }
    bc = 100.0f / fmaxf(sqrtf(s), 1e-8f);   // BETA folded into normalized key
  }
  __syncthreads();
  if (t < DD) kn[t] = key[b*DD + t] * bc;
  __syncthreads();

  // pass 1: sim = beta * k_n . M_n, track max
  float lmax = -3.402823466e38f;
  for (int n = t; n < NN; n += 256) {
    const float* row = Mf + n * DD;
    float s = 0.f;
    #pragma unroll
    for (int dd = 0; dd < DD; ++dd) s += kn[dd] * row[dd];
    s *= invnorm[n];
    sim[b*NN + n] = s;
    lmax = fmaxf(lmax, s);
  }
  #pragma unroll
  for (int off = 16; off > 0; off >>= 1) lmax = fmaxf(lmax, __shfl_down(lmax, off));
  if ((t & 31) == 0) red[t >> 5] = lmax;
  __syncthreads();
  if (t == 0) {
    float mm = red[0];
    for (int i = 1; i < 8; ++i) mm = fmaxf(mm, red[i]);
    bc = mm;
  }
  __syncthreads();
  const float gmax = bc;
  __syncthreads();

  // pass 2: exp + sum
  float lsum = 0.f;
  for (int n = t; n < NN; n += 256) {
    const float ev = expf(sim[b*NN + n] - gmax);
    sim[b*NN + n] = ev;
    lsum += ev;
  }
  #pragma unroll
  for (int off = 16; off > 0; off >>= 1) lsum += __shfl_down(lsum, off);
  if ((t & 31) == 0) red[t >> 5] = lsum;
  __syncthreads();
  if (t == 0) {
    float s = 0.f;
    for (int i = 0; i < 8; ++i) s += red[i];
    bc = 1.0f / fmaxf(s, 1e-12f);           // softmax + L1 renorm (gamma=1)
  }
  __syncthreads();
  const float invs = bc;
  for (int n = t; n < NN; n += 256) w[b*NN + n] = sim[b*NN + n] * invs;
}

// ---------------------------------------------------------------------------
// K3: rank-update corrections  S1 = w @ w_t^T,  S2 = w @ w_prev^T  (32x32).
// grid (2, 32): x = which matrix, y = i. Block 1024: wave wv owns j=wv,
// lanes stride n (coalesced), shuffle-reduce.
// ---------------------------------------------------------------------------
__global__ __launch_bounds__(1024) void k_corr(
    const float* __restrict__ w, const float* __restrict__ wt,
    const float* __restrict__ wprev, float* __restrict__ S1,
    float* __restrict__ S2)
{
  const int lane = threadIdx.x & 31;
  const int j    = threadIdx.x >> 5;   // 0..31
  const int i    = blockIdx.y;         // 0..31
  const float* V = (blockIdx.x == 0) ? wt : wprev;
  float* S       = (blockIdx.x == 0) ? S1 : S2;
  float s = 0.f;
  for (int n = lane; n < NN; n += 32)
    s += w[(size_t)i * NN + n] * V[(size_t)j * NN + n];
  #pragma unroll
  for (int off = 16; off > 0; off >>= 1) s += __shfl_down(s, off);
  if (lane == 0) S[i * 32 + j] = s;
}

// ---------------------------------------------------------------------------
// K4: the heavy pass.  dir==0: t_fwd = w@E + S1@w_prev
//                      dir==1: t_bwd = w@E^T + S2@w_t
// Block = 8 waves = 2 column tiles x 4 K-sections (KSPLIT). Each wave runs
// two 16x16x(K=2048) WMMA f32 chains (batch rows 0-15 / 16-31), then K-section
// partials are combined through LDS; section-0 waves add the 32x32 rank-update
// correction and store. Fragment layouts per ISA 16x16x4 f32:
//   A lane L -> {A[L%16][kh], A[L%16][kh+1]}, kh = (L>=16)?2:0
//   B lane L -> {B[kh][L%16], B[kh+1][L%16]};  C/D vgpr i -> rows i / i+8.
// ---------------------------------------------------------------------------
__global__ __launch_bounds__(256) void k_gemm_wmma(
    const float* __restrict__ E, const float* __restrict__ w,
    const float* __restrict__ wt, const float* __restrict__ wprev,
    const float* __restrict__ S1, const float* __restrict__ S2,
    float* __restrict__ tfwd, float* __restrict__ tbwd)
{
  const int lane = threadIdx.x & 31;
  const int wv   = threadIdx.x >> 5;   // 0..7
  const int tile = wv & 1;             // 0..1
  const int ksec = wv >> 1;            // 0..3
  const int dir  = blockIdx.y;
  const int c0   = blockIdx.x * 32 + tile * 16;   // 16 output columns per wave
  const int mrow = lane & 15;
  const int kh   = (lane >> 4) << 1;   // 0 or 2
  const int kbeg = ksec * KCH;

  v8f acc0 = {0.f,0.f,0.f,0.f,0.f,0.f,0.f,0.f};
  v8f acc1 = {0.f,0.f,0.f,0.f,0.f,0.f,0.f,0.f};

  const float* wA0 = w + (size_t)mrow * NN + kh;          // batch rows 0-15
  const float* wA1 = w + (size_t)(mrow + 16) * NN + kh;   // batch rows 16-31

  if (dir == 0) {
    // B[k][n] = E[k][c0+n]  (column strip of E)
    for (int k0 = kbeg; k0 < kbeg + KCH; k0 += 4) {
      v2f a0, a1, bf;
      a0.x = wA0[k0]; a0.y = wA0[k0 + 1];
      a1.x = wA1[k0]; a1.y = wA1[k0 + 1];
      const size_t be = (size_t)(k0 + kh) * NN + (size_t)(c0 + mrow);
      bf.x = E[be];
      bf.y = E[be + NN];
      acc0 = __builtin_amdgcn_wmma_f32_16x16x4_f32(false, a0, false, bf,
                                                   (short)0, acc0, false, false);
      acc1 = __builtin_amdgcn_wmma_f32_16x16x4_f32(false, a1, false, bf,
                                                   (short)0, acc1, false, false);
    }
  } else {
    // B[k][n] = E[c0+n][k]  (row strip of E, contiguous float2 per lane)
    const float* Erow = E + (size_t)(c0 + mrow) * NN + kh;
    for (int k0 = kbeg; k0 < kbeg + KCH; k0 += 4) {
      v2f a0, a1, bf;
      a0.x = wA0[k0]; a0.y = wA0[k0 + 1];
      a1.x = wA1[k0]; a1.y = wA1[k0 + 1];
      bf.x = Erow[k0]; bf.y = Erow[k0 + 1];
      acc0 = __builtin_amdgcn_wmma_f32_16x16x4_f32(false, a0, false, bf,
                                                   (short)0, acc0, false, false);
      acc1 = __builtin_amdgcn_wmma_f32_16x16x4_f32(false, a1, false, bf,
                                                   (short)0, acc1, false, false);
    }
  }

  // combine K-section partials through LDS
  __shared__ float sh[8 * 32 * 16];    // 16 KB
  float* my = sh + ((size_t)threadIdx.x << 4);
  #pragma unroll
  for (int i = 0; i < 8; ++i) { my[i] = acc0[i]; my[i + 8] = acc1[i]; }
  __syncthreads();

  if (ksec == 0) {
    #pragma unroll
    for (int s = 1; s < KSPLIT; ++s) {
      const float* p = sh + (((s * 2 + tile) * 32 + lane) << 4);
      #pragma unroll
      for (int i = 0; i < 8; ++i) { acc0[i] += p[i]; acc1[i] += p[i + 8]; }
    }

    // epilogue: add 32x32 rank-update correction, write out
    const float* S = dir ? S2 : S1;
    const float* V = dir ? wt : wprev;
    float* T       = dir ? tbwd : tfwd;
    const int nn    = c0 + mrow;
    const int mbase = (lane >> 4) << 3;            // 0 or 8 per C/D layout
    #pragma unroll
    for (int i = 0; i < 8; ++i) {
      const int m = i + mbase;
      float s0 = 0.f, s1 = 0.f;
      for (int j = 0; j < 32; ++j) {
        const float vj = V[(size_t)j * NN + nn];
        s0 += S[m * 32 + j] * vj;
        s1 += S[(m + 16) * 32 + j] * vj;
      }
      T[(size_t)m * NN + nn]        = acc0[i] + s0;
      T[(size_t)(m + 16) * NN + nn] = acc1[i] + s1;
    }
  }
}

// ---------------------------------------------------------------------------
// K5: final readout  r_normal = w@Mf, r_forward = tfwd@Mf, r_backward = tbwd@Mf
// grid (3, 32), 256 threads: 4 n-chunks x 64 d, 4 accumulators per thread,
// LDS combine across chunks.
// ---------------------------------------------------------------------------
__global__ __launch_bounds__(256) void k_readout(
    const float* __restrict__ w, const float* __restrict__ tf,
    const float* __restrict__ tb, const float* __restrict__ Mf,
    float* __restrict__ out)
{
  const int mat = blockIdx.x;
  const int b   = blockIdx.y;
  const float* coef = (mat == 0) ? w : ((mat == 1) ? tf : tb);
  const int d  = threadIdx.x & 63;
  const int ch = threadIdx.x >> 6;                // 0..3
  const int nb = ch * (NN / 4);
  float s0 = 0.f, s1 = 0.f, s2 = 0.f, s3 = 0.f;
  for (int n = nb; n < nb + NN / 4; n += 4) {
    s0 += coef[b*NN + n + 0] * Mf[(n + 0) * DD + d];
    s1 += coef[b*NN + n + 1] * Mf[(n + 1) * DD + d];
    s2 += coef[b*NN + n + 2] * Mf[(n + 2) * DD + d];
    s3 += coef[b*NN + n + 3] * Mf[(n + 3) * DD + d];
  }
  __shared__ float sh[256];
  sh[threadIdx.x] = (s0 + s1) + (s2 + s3);
  __syncthreads();
  if (ch == 0) {
    const float s = (sh[d] + sh[64 + d]) + (sh[128 + d] + sh[192 + d]);
    out[mat * (BB * DD) + b * DD + d] = s;
  }
}

// ---------------------------------------------------------------------------
extern "C" void kernel_launch(void* const* d_in, const int* in_sizes, int n_in,
                              void* d_out, int out_size, void* d_ws, size_t ws_size,
                              hipStream_t stream) {
  const float* wt    = (const float*)d_in[0];  // [B,N]
  const float* wprev = (const float*)d_in[1];  // [B,N]
  const float* e     = (const float*)d_in[2];  // [B,D]
  const float* a     = (const float*)d_in[3];  // [B,D]
  const float* key   = (const float*)d_in[4];  // [B,D]
  const float* M     = (const float*)d_in[5];  // [N,D]
  const float* E     = (const float*)d_in[6];  // [N,N]
  float* out = (float*)d_out;

  float* ws      = (float*)d_ws;
  float* Mf      = ws;                 // N*D
  float* invnorm = Mf + NN * DD;       // N
  float* w       = invnorm + NN;       // B*N
  float* sim     = w + BB * NN;        // B*N (scratch)
  float* S1      = sim + BB * NN;      // 32*32
  float* S2      = S1 + 1024;          // 32*32
  float* tfwd    = S2 + 1024;          // B*N
  float* tbwd    = tfwd + BB * NN;     // B*N   (total ~6.3 MB)

  k_write_scan<<<dim3(NN), dim3(DD), 0, stream>>>(wt, e, a, M, Mf, invnorm);
  k_address  <<<dim3(BB), dim3(256), 0, stream>>>(key, Mf, invnorm, sim, w);
  k_corr     <<<dim3(2, BB), dim3(1024), 0, stream>>>(w, wt, wprev, S1, S2);
  k_gemm_wmma<<<dim3(NN / 32, 2), dim3(256), 0, stream>>>(E, w, wt, wprev,
                                                          S1, S2, tfwd, tbwd);
  k_readout  <<<dim3(3, BB), dim3(256), 0, stream>>>(w, tfwd, tbwd, Mf, out);
}